// GlobalNormRLE_52578989637854
// MI455X (gfx1250) — compile-verified
//
#include <hip/hip_runtime.h>
#include <hip/hip_bf16.h>

// Problem constants (from reference setup_inputs)
#define TT     2048
#define BATCH  128
#define INSZ   512
#define SZ     16
#define NBASE  4
#define NCHUNK 32
#define CHLEN  (TT / NCHUNK)      // 64
#define WPITCH 516                // padded LDS row pitch (floats) -> conflict-free ds_load_b64

typedef __attribute__((ext_vector_type(2))) float v2f;
typedef __attribute__((ext_vector_type(8))) float v8f;

__device__ __forceinline__ float logaddexpf_(float a, float b) {
    float mx = fmaxf(a, b);
    float mn = fminf(a, b);
    // if mn == -inf this returns mx exactly (exp(-inf)=0)
    return mx + log1pf(expf(mn - mx));
}

// ---------------------------------------------------------------------------
// Phase 1: y = x @ W^T + b via V_WMMA_F32_16X16X4_F32, fused activations.
// One wave computes a 16-row x 16-output tile. Block = 8 waves = 8 tiles.
// A (16x4 f32): lane<16 holds M=lane, K={0,1}; lane>=16 holds M=lane-16, K={2,3}
// B (4x16 f32): lane<16 holds N=lane, K={0,1}; lane>=16 holds N=lane-16, K={2,3}
//   with B[k][n] = W[n][k]  ->  same addressing as A but indexed into W rows.
// C (16x16 f32): vgpr r, lane<16 -> (M=r, N=lane); lane>=16 -> (M=8+r, N=lane-16)
// ---------------------------------------------------------------------------
__global__ __launch_bounds__(256) void p1_gemm_act(
    const float* __restrict__ x, const float* __restrict__ W,
    const float* __restrict__ bias, float* __restrict__ out)
{
    __shared__ float Wlds[SZ * WPITCH];            // ~33 KB of 320 KB

    // Cooperative, coalesced stage of W (16 x 512) into padded LDS.
    for (int i = threadIdx.x; i < SZ * INSZ; i += 256) {
        int r = i >> 9;          // /512
        int c = i & (INSZ - 1);
        Wlds[r * WPITCH + c] = W[i];
    }
    __syncthreads();

    const int wave = threadIdx.x >> 5;
    const int lane = threadIdx.x & 31;
    const int lo   = lane & 15;
    const int hi   = lane >> 4;

    const long tile    = (long)blockIdx.x * 8 + wave;
    const long baseRow = tile * 16;

    const float* xrow = x + (baseRow + lo) * INSZ + 2 * hi;
    const float* wrow = Wlds + lo * WPITCH + 2 * hi;

    v8f acc = {};
#pragma unroll 8
    for (int k = 0; k < INSZ; k += 4) {
        v2f a  = *(const v2f*)(xrow + k);   // global, L0-friendly streaming
        v2f bb = *(const v2f*)(wrow + k);   // LDS, conflict-free
        acc = __builtin_amdgcn_wmma_f32_16x16x4_f32(
            /*neg_a=*/false, a, /*neg_b=*/false, bb,
            /*c_mod=*/(short)0, acc, /*reuse_a=*/false, /*reuse_b=*/false);
    }

    const float bv = bias[lo];   // output channel s == lo for every acc register
#pragma unroll
    for (int r = 0; r < 8; ++r) {
        float yv = acc[r] + bv;
        float sp = fmaxf(yv, 0.0f) + log1pf(expf(-fabsf(yv)));  // stable softplus
        float res = (lo < NBASE)     ? (1.0f + sp)
                  : (lo < 2 * NBASE) ? (0.1f + sp)
                                     : (5.0f * tanhf(yv));
        out[(baseRow + r + 8 * hi) * SZ + lo] = res;
    }
}

// ---------------------------------------------------------------------------
// Phase 2a: per-chunk affine map in log-space.
// Exp-space step: v <- A_t v + c_t,  A_t[s,s]=e^stay[s], A_t[s,s']=e^move[s],
// c_t[s]=e^move[s].  Thread = (batch, column); columns 0..3 = matrix columns,
// column 4 = affine column (gets the extra +c_t term).
// ---------------------------------------------------------------------------
__global__ __launch_bounds__(640) void p2_chunk(
    const float* __restrict__ out, float* __restrict__ maps)
{
    const int chunk = blockIdx.x;
    const int b   = threadIdx.x & (BATCH - 1);
    const int col = threadIdx.x >> 7;            // 0..4
    const int t0  = chunk * CHLEN;

    const float* tr = out + ((long)t0 * BATCH + b) * SZ + 2 * NBASE;  // trans[t0,b,:]
    const long tstride = (long)BATCH * SZ;

    float4 mv = *(const float4*)(tr);        // move = trans[..., 0:4]
    float4 st = *(const float4*)(tr + 4);    // stay = trans[..., 4:8]

    // Analytic first step (apply A_{t0} / c_{t0} to identity map).
    float c0, c1, c2, c3;
    if (col == 4) { c0 = mv.x; c1 = mv.y; c2 = mv.z; c3 = mv.w; }
    else {
        c0 = (col == 0) ? st.x : mv.x;
        c1 = (col == 1) ? st.y : mv.y;
        c2 = (col == 2) ? st.z : mv.z;
        c3 = (col == 3) ? st.w : mv.w;
    }

    for (int t = 1; t < CHLEN; ++t) {
        const float* p = tr + (long)t * tstride;
        mv = *(const float4*)(p);
        st = *(const float4*)(p + 4);

        float m  = fmaxf(fmaxf(c0, c1), fmaxf(c2, c3));
        float e0 = expf(c0 - m), e1 = expf(c1 - m);
        float e2 = expf(c2 - m), e3 = expf(c3 - m);
        // direct off-diagonal sums: no cancellation
        float r0 = e1 + e2 + e3, r1 = e0 + e2 + e3;
        float r2 = e0 + e1 + e3, r3 = e0 + e1 + e2;

        float n0 = logaddexpf_(st.x + c0, mv.x + m + logf(r0));
        float n1 = logaddexpf_(st.y + c1, mv.y + m + logf(r1));
        float n2 = logaddexpf_(st.z + c2, mv.z + m + logf(r2));
        float n3 = logaddexpf_(st.w + c3, mv.w + m + logf(r3));
        if (col == 4) {  // affine column: + c_t
            n0 = logaddexpf_(n0, mv.x);
            n1 = logaddexpf_(n1, mv.y);
            n2 = logaddexpf_(n2, mv.z);
            n3 = logaddexpf_(n3, mv.w);
        }
        c0 = n0; c1 = n1; c2 = n2; c3 = n3;
    }

    float* dst = maps + (((long)chunk * BATCH + b) * 5 + col) * 4;
    dst[0] = c0; dst[1] = c1; dst[2] = c2; dst[3] = c3;
}

// ---------------------------------------------------------------------------
// Phase 2b: compose the 32 chunk maps in order, applied to fwd0 = zeros(log).
// One lane per batch.  lz[b] = logsumexp(fwd)/T.
// ---------------------------------------------------------------------------
__global__ __launch_bounds__(128) void p2_combine(
    const float* __restrict__ maps, float* __restrict__ lz)
{
    const int b = threadIdx.x;
    float f0 = 0.0f, f1 = 0.0f, f2 = 0.0f, f3 = 0.0f;  // log(ones) = 0

    for (int k = 0; k < NCHUNK; ++k) {
        const float* mp = maps + ((long)k * BATCH + b) * 20;
        float n[4];
#pragma unroll
        for (int s = 0; s < 4; ++s) {
            float a0 = mp[0 * 4 + s] + f0;
            float a1 = mp[1 * 4 + s] + f1;
            float a2 = mp[2 * 4 + s] + f2;
            float a3 = mp[3 * 4 + s] + f3;
            float m  = fmaxf(fmaxf(a0, a1), fmaxf(a2, a3));
            float sm = expf(a0 - m) + expf(a1 - m) + expf(a2 - m) + expf(a3 - m);
            n[s] = logaddexpf_(m + logf(sm), mp[16 + s]);   // + affine column
        }
        f0 = n[0]; f1 = n[1]; f2 = n[2]; f3 = n[3];
    }
    float m  = fmaxf(fmaxf(f0, f1), fmaxf(f2, f3));
    float sZ = expf(f0 - m) + expf(f1 - m) + expf(f2 - m) + expf(f3 - m);
    lz[b] = (m + logf(sZ)) / (float)TT;
}

// ---------------------------------------------------------------------------
// Phase 2c: trans -= logZ[b]/T, in place on d_out (one row of 8 per thread).
// ---------------------------------------------------------------------------
__global__ __launch_bounds__(256) void p2_apply(
    float* __restrict__ out, const float* __restrict__ lz)
{
    const long row = (long)blockIdx.x * 256 + threadIdx.x;
    const int  b   = (int)(row & (BATCH - 1));
    const float z  = lz[b];
    float4* p = (float4*)(out + row * SZ + 2 * NBASE);
    float4 a = p[0], c = p[1];
    a.x -= z; a.y -= z; a.z -= z; a.w -= z;
    c.x -= z; c.y -= z; c.z -= z; c.w -= z;
    p[0] = a; p[1] = c;
}

extern "C" void kernel_launch(void* const* d_in, const int* in_sizes, int n_in,
                              void* d_out, int out_size, void* d_ws, size_t ws_size,
                              hipStream_t stream)
{
    const float* x    = (const float*)d_in[0];   // (2048,128,512) f32
    const float* W    = (const float*)d_in[1];   // (16,512) f32
    const float* bias = (const float*)d_in[2];   // (16,) f32
    float* out = (float*)d_out;                  // (2048,128,16) f32

    float* maps = (float*)d_ws;                          // 32*128*20 floats = 320 KB
    float* lz   = maps + (long)NCHUNK * BATCH * 20;      // 128 floats

    const int tiles  = (TT * BATCH) / 16;                // 16384
    const int blocks = tiles / 8;                        // 2048 (8 waves/tiles per block)

    p1_gemm_act<<<blocks, 256, 0, stream>>>(x, W, bias, out);
    p2_chunk  <<<NCHUNK, 640, 0, stream>>>(out, maps);
    p2_combine<<<1, 128, 0, stream>>>(maps, lz);
    p2_apply  <<<(TT * BATCH) / 256, 256, 0, stream>>>(out, lz);
}